// PixAdaConvNet_13881334301299
// MI455X (gfx1250) — compile-verified
//
#include <hip/hip_runtime.h>
#include <hip/hip_bf16.h>

typedef float v2f __attribute__((ext_vector_type(2)));
typedef float v8f __attribute__((ext_vector_type(8)));

#define B_      2
#define C_      3
#define H_      160
#define W_      160
#define HW_     (H_ * W_)        // 25600
#define L_      72
#define NCODE   3000
#define S_      4
#define K_      5
#define M_TOT   (B_ * HW_)       // 51200 pixels
#define NTILE   64
#define NT_     ((NCODE + NTILE - 1) / NTILE)   // 47 N tiles
#define KSTEPS  (L_ / 4)         // 18 WMMA K-steps
#define LDS_STRIDE 74            // even (8B-aligned float2), 74%64=10 -> conflict-free
#define BUFSZ   (NTILE * LDS_STRIDE)
#define STG     ((NTILE * L_) / 256)            // 18 staged elements / thread

// ---------------------------------------------------------------------------
// Kernel 1: fused routing GEMM + argmax, double-buffered keys staging.
//   sim[m, n] = sum_l Q[m, l] * keys[n, l],  idx[m] = argmax_n sim[m, n]
// One wave owns a 16-row M stripe; A (16x72) lives in registers for the whole
// N loop. keys staged per 64-row N tile into LDS (2 buffers), shared by 8
// waves; global loads for tile t+1 overlap the 72 WMMAs of tile t.
// f32 WMMA (V_WMMA_F32_16X16X4_F32) => exact reference math on matrix pipes.
// ---------------------------------------------------------------------------
__global__ __launch_bounds__(256)
void route_argmax_kernel(const float* __restrict__ q,     // [B, L, H, W]
                         const float* __restrict__ keys,  // [NCODE, L]
                         int* __restrict__ idx_out)       // [M_TOT]
{
    __shared__ float klds[2 * BUFSZ];

    const int tid  = threadIdx.x;
    const int lane = tid & 31;
    const int wave = tid >> 5;
    const int half = lane >> 4;   // 0: rows 0-7 / K sel +0 ; 1: rows 8-15 / +2
    const int l15  = lane & 15;

    const int mbase = blockIdx.x * 128 + wave * 16;

    // ---- Preload A stripe: 16(M) x 72(K), per the 32-bit A 16x4 lane layout.
    // lanes 0-15 hold K = 4kk+{0,1}, lanes 16-31 hold K = 4kk+{2,3}, M = l15.
    const int m  = mbase + l15;
    const int b  = m / HW_;
    const int hw = m - b * HW_;
    const float* qb = q + (size_t)b * L_ * HW_ + hw;   // element (b, l, hw) at l*HW
    v2f afrag[KSTEPS];
#pragma unroll
    for (int kk = 0; kk < KSTEPS; ++kk) {
        const int k0 = kk * 4 + half * 2;
        afrag[kk].x = qb[(size_t)k0 * HW_];
        afrag[kk].y = qb[(size_t)(k0 + 1) * HW_];
    }

    // Per-lane running argmax: entry j tracks row M = j + 8*half over the
    // N-residue class this lane owns (N % 16 == l15).
    float bestV[8];
    int   bestI[8];
#pragma unroll
    for (int j = 0; j < 8; ++j) { bestV[j] = -3.402823466e38f; bestI[j] = 0; }

    const v8f vzero = {0.f, 0.f, 0.f, 0.f, 0.f, 0.f, 0.f, 0.f};

    // ---- Stage tile 0 into buffer 0.
    float stage[STG];
#pragma unroll
    for (int t = 0; t < STG; ++t) {
        const int e = t * 256 + tid;
        const int r = e / L_, cc = e - (e / L_) * L_;
        stage[t] = (r < NCODE) ? keys[(size_t)r * L_ + cc] : 0.0f;
    }
#pragma unroll
    for (int t = 0; t < STG; ++t) {
        const int e = t * 256 + tid;
        const int r = e / L_, cc = e - (e / L_) * L_;
        klds[r * LDS_STRIDE + cc] = stage[t];
    }
    __syncthreads();

    int cur = 0;
    for (int nt = 0; nt < NT_; ++nt) {
        const int nbase = nt * NTILE;

        // ---- Issue global loads for tile nt+1 (latency hides under WMMAs).
        if (nt + 1 < NT_) {
            const int nb = nbase + NTILE;
#pragma unroll
            for (int t = 0; t < STG; ++t) {
                const int e  = t * 256 + tid;
                const int r  = e / L_, cc = e - (e / L_) * L_;
                const int ng = nb + r;
                stage[t] = (ng < NCODE) ? keys[(size_t)ng * L_ + cc] : 0.0f;
            }
            // Warm L2 two tiles ahead (global_prefetch_b8).
            if (tid < NTILE && (nbase + 2 * NTILE) < NCODE)
                __builtin_prefetch(&keys[(size_t)(nbase + 2 * NTILE + tid) * L_], 0, 1);
        }

        // ---- 16x64 tile: 4 N-subtiles, 18 K-steps each, interleaved for ILP.
        const float* bufc = klds + cur * BUFSZ;
        v8f acc[4];
#pragma unroll
        for (int t = 0; t < 4; ++t) acc[t] = vzero;

#pragma unroll
        for (int kk = 0; kk < KSTEPS; ++kk) {
            const int k0 = kk * 4 + half * 2;
#pragma unroll
            for (int t = 0; t < 4; ++t) {
                // B 4x16 fragment: lane = column N, K selection mirrors A.
                const v2f bfrag =
                    *(const v2f*)&bufc[(l15 + 16 * t) * LDS_STRIDE + k0];
                acc[t] = __builtin_amdgcn_wmma_f32_16x16x4_f32(
                    /*neg_a=*/false, afrag[kk],
                    /*neg_b=*/false, bfrag,
                    /*c_mod=*/(short)0, acc[t],
                    /*reuse_a=*/false, /*reuse_b=*/false);
            }
        }

        // ---- Fold tile into the running argmax (strict '>' keeps first max).
#pragma unroll
        for (int t = 0; t < 4; ++t) {
            const int  ng    = nbase + t * 16 + l15;
            const bool valid = (ng < NCODE);
#pragma unroll
            for (int j = 0; j < 8; ++j) {
                const float v = acc[t][j];
                if (valid && v > bestV[j]) { bestV[j] = v; bestI[j] = ng; }
            }
        }

        // ---- Drain staged loads into the other buffer, barrier, swap.
        if (nt + 1 < NT_) {
            float* bufn = klds + (1 - cur) * BUFSZ;
#pragma unroll
            for (int t = 0; t < STG; ++t) {
                const int e = t * 256 + tid;
                const int r = e / L_, cc = e - (e / L_) * L_;
                bufn[r * LDS_STRIDE + cc] = stage[t];
            }
        }
        __syncthreads();
        cur ^= 1;
    }

    // ---- Cross-lane argmax within each 16-lane half (columns of the tile).
#pragma unroll
    for (int off = 1; off < 16; off <<= 1) {
#pragma unroll
        for (int j = 0; j < 8; ++j) {
            const float ov = __shfl_xor(bestV[j], off, 32);
            const int   oi = __shfl_xor(bestI[j], off, 32);
            if (ov > bestV[j] || (ov == bestV[j] && oi < bestI[j])) {
                bestV[j] = ov; bestI[j] = oi;
            }
        }
    }
    if (l15 == 0) {
#pragma unroll
        for (int j = 0; j < 8; ++j)
            idx_out[mbase + half * 8 + j] = bestI[j];
    }
}

// ---------------------------------------------------------------------------
// Kernel 2: gather routed kernels + reflect-padded 5x5 dynamic conv + shuffle.
// One thread per OUTPUT pixel -> fully coalesced stores. ~123 MFLOP total.
// ---------------------------------------------------------------------------
__global__ __launch_bounds__(256)
void dynconv_kernel(const float* __restrict__ x,       // [B, C, H, W]
                    const float* __restrict__ values,  // [NCODE, 16, 25]
                    const int*   __restrict__ idx,     // [M_TOT]
                    float* __restrict__ out)           // [B, C, 4H, 4W]
{
    const int o  = blockIdx.x * 256 + threadIdx.x;
    const int WO = W_ * S_;                 // 640
    const int HO = H_ * S_;                 // 640

    const int xo = o % WO;
    int t        = o / WO;
    const int y  = t % HO;
    t           /= HO;
    const int c  = t % C_;
    const int b  = t / C_;

    const int w  = xo >> 2, sj = xo & 3;
    const int h  = y  >> 2, si = y  & 3;

    const int n  = idx[(b * H_ + h) * W_ + w];
    const float* vr = values + ((size_t)n * (S_ * S_) + (si * S_ + sj)) * (K_ * K_);
    const float* xb = x + (size_t)(b * C_ + c) * HW_;

    float acc = 0.0f;
#pragma unroll
    for (int dy = 0; dy < K_; ++dy) {
        int yy = h + dy - 2;
        yy = yy < 0 ? -yy : (yy >= H_ ? 2 * H_ - 2 - yy : yy);   // reflect
#pragma unroll
        for (int dx = 0; dx < K_; ++dx) {
            int xx = w + dx - 2;
            xx = xx < 0 ? -xx : (xx >= W_ ? 2 * W_ - 2 - xx : xx);
            acc += xb[yy * W_ + xx] * vr[dy * K_ + dx];
        }
    }
    out[o] = acc;
}

// ---------------------------------------------------------------------------
extern "C" void kernel_launch(void* const* d_in, const int* in_sizes, int n_in,
                              void* d_out, int out_size, void* d_ws, size_t ws_size,
                              hipStream_t stream) {
    const float* x       = (const float*)d_in[0];  // [2,3,160,160]
    const float* queries = (const float*)d_in[1];  // [2,72,160,160]
    const float* keys    = (const float*)d_in[2];  // [3000,72]
    const float* values  = (const float*)d_in[3];  // [3000,16,25]
    (void)in_sizes; (void)n_in; (void)ws_size;

    int* idx = (int*)d_ws;                         // 51200 ints of scratch

    // Phase 1: fused routing GEMM + argmax (WMMA f32), 400 WGs x 8 waves.
    route_argmax_kernel<<<M_TOT / 128, 256, 0, stream>>>(queries, keys, idx);

    // Phase 2: dynamic conv + pixel shuffle, one thread per output element.
    const int nout = B_ * C_ * (H_ * S_) * (W_ * S_);   // 2,457,600
    dynconv_kernel<<<nout / 256, 256, 0, stream>>>(x, values, idx, (float*)d_out);
}